// UNet_block_SA_inter_intra_blockWidth_Stride_Kernel_second_layer_33105607917989
// MI455X (gfx1250) — compile-verified
//
#include <hip/hip_runtime.h>

typedef __attribute__((ext_vector_type(2))) float v2f;
typedef __attribute__((ext_vector_type(8))) float v8f;

#define HW   128
#define IMG  (HW * HW)     // 16384
#define NCH  64
#define NSTR 2304          // max window N (48*48), used as fixed stride
#define QOWN 1024          // owned queries per window (32*32)

__device__ __forceinline__ v8f wmma_f32(v2f a, v2f b, v8f c) {
  // D = A(16x4, f32) * B(4x16, f32) + C(16x16, f32)
  return __builtin_amdgcn_wmma_f32_16x16x4_f32(false, a, false, b, (short)0, c,
                                               false, false);
}

__device__ __forceinline__ float rmax16(float v) {
  v = fmaxf(v, __shfl_xor(v, 1, 32));
  v = fmaxf(v, __shfl_xor(v, 2, 32));
  v = fmaxf(v, __shfl_xor(v, 4, 32));
  v = fmaxf(v, __shfl_xor(v, 8, 32));
  return v;
}
__device__ __forceinline__ float rsum16(float v) {
  v += __shfl_xor(v, 1, 32);
  v += __shfl_xor(v, 2, 32);
  v += __shfl_xor(v, 4, 32);
  v += __shfl_xor(v, 8, 32);
  return v;
}

// ---------------------------------------------------------------------------
// QKV kernel: one wave per (window, 16-position tile).
//   q[oi][c]  (owned 32x32 block only)   k[n][c]   v[c][n]
// GEMM: D(16 out-ch x 16 pos) = W(16x64) * X(64x16), 16 WMMA k-steps.
// ---------------------------------------------------------------------------
__global__ __launch_bounds__(32) void qkv_kernel(
    const float* __restrict__ src,  // [C][HW][HW] for this batch
    const float* __restrict__ Wq, const float* __restrict__ Wk,
    const float* __restrict__ Wv, const float* __restrict__ bq,
    const float* __restrict__ bk, const float* __restrict__ bv,
    float* __restrict__ qbuf, float* __restrict__ kbuf,
    float* __restrict__ vbuf) {
  const int win  = blockIdx.x / 144;
  const int tile = blockIdx.x % 144;
  const int wi = win >> 2, wj = win & 3;
  const int wh = (wi == 3) ? 32 : 48;
  const int ww = (wj == 3) ? 32 : 48;
  const int N = wh * ww;
  if (tile * 16 >= N) return;  // uniform per wave -> EXEC stays all-ones

  const int L   = threadIdx.x;
  const int l16 = L & 15;
  const int hh  = L >> 4;  // half-wave selects the K pair inside a WMMA step

  const int n  = tile * 16 + l16;
  const int lr = n / ww;
  const int lc = n - lr * ww;
  const int soff = (wi * 32 + lr) * HW + (wj * 32 + lc);

  // B operand (x tile) for all 16 k-steps: B[k=c'][n], lane n = l16
  float bx[16], by[16];
#pragma unroll
  for (int kk = 0; kk < 16; ++kk) {
    const int c0 = kk * 4 + 2 * hh;
    bx[kk] = src[c0 * IMG + soff];
    by[kk] = src[(c0 + 1) * IMG + soff];
  }

  const float* Ws[3] = {Wq, Wk, Wv};
  const float* Bs[3] = {bq, bk, bv};

  for (int comp = 0; comp < 3; ++comp) {
    const float* W  = Ws[comp];
    const float* Bb = Bs[comp];
#pragma unroll
    for (int ct = 0; ct < 4; ++ct) {
      v8f acc = {};
#pragma unroll
      for (int kk = 0; kk < 16; ++kk) {
        v2f a = *(const v2f*)(W + (ct * 16 + l16) * NCH + kk * 4 + 2 * hh);
        v2f b;
        b.x = bx[kk];
        b.y = by[kk];
        acc = wmma_f32(a, b, acc);
      }
      // D layout: VGPR r -> out-channel d = ct*16 + r + 8*hh, position n
#pragma unroll
      for (int r = 0; r < 8; ++r) {
        const int d   = ct * 16 + r + 8 * hh;
        const float v = acc[r] + Bb[d];
        if (comp == 1) {  // k : [n][64]
          kbuf[(size_t)(win * NSTR + n) * NCH + d] = v;
        } else if (comp == 2) {  // v : [64][NSTR]
          vbuf[(size_t)win * NSTR * NCH + (size_t)d * NSTR + n] = v;
        } else {  // q : owned positions only, [oi][64]
          if (lr < 32 && lc < 32)
            qbuf[(size_t)(win * QOWN + lr * 32 + lc) * NCH + d] = v;
        }
      }
    }
  }
}

// ---------------------------------------------------------------------------
// Flash-attention kernel: one wave per (window, 16-query tile). 8 waves/block.
// ---------------------------------------------------------------------------
__global__ __launch_bounds__(256) void attn_kernel(
    const float* __restrict__ qbuf, const float* __restrict__ kbuf,
    const float* __restrict__ vbuf, const float* __restrict__ src,
    float* __restrict__ dst) {
  __shared__ float plds[8 * 16 * 20];  // per-wave padded 16x16 P staging
  const int L    = threadIdx.x & 31;
  const int wid  = threadIdx.x >> 5;
  const int task = blockIdx.x * 8 + wid;  // 0..1023
  const int win  = task >> 6;
  const int qt   = task & 63;
  const int wi = win >> 2, wj = win & 3;
  const int wh  = (wi == 3) ? 32 : 48;
  const int ww  = (wj == 3) ? 32 : 48;
  const int nkt = (wh * ww) >> 4;  // 64..144 key tiles (uniform per wave)

  const int l16 = L & 15;
  const int hh  = L >> 4;
  float* myp = plds + wid * 320;

  // A operand (q^T): rows m = l16 -> query oi = qt*16 + l16, [m][c] contiguous
  v2f aq[16];
  const float* qb = qbuf + (size_t)(win * QOWN + qt * 16 + l16) * NCH;
#pragma unroll
  for (int kk = 0; kk < 16; ++kk)
    aq[kk] = *(const v2f*)(qb + kk * 4 + 2 * hh);

  v8f acc[4];
  float mrow[8], lrow[8];
#pragma unroll
  for (int ct = 0; ct < 4; ++ct) acc[ct] = (v8f){};
#pragma unroll
  for (int r = 0; r < 8; ++r) {
    mrow[r] = -__builtin_inff();
    lrow[r] = 0.f;
  }

  const float* kb = kbuf + (size_t)win * NSTR * NCH;
  const float* vb = vbuf + (size_t)win * NSTR * NCH;

  for (int jt = 0; jt < nkt; ++jt) {
    // E = q^T k for this 16x16 tile
    v8f E = {};
    const float* kt = kb + (size_t)(jt * 16 + l16) * NCH;
#pragma unroll
    for (int kk = 0; kk < 16; ++kk) {
      v2f b = *(const v2f*)(kt + kk * 4 + 2 * hh);
      E = wmma_f32(aq[kk], b, E);
    }
    // online softmax; row stats live in the correct lanes per half-wave
    float scr[8];
#pragma unroll
    for (int r = 0; r < 8; ++r) {
      const float rm = rmax16(E[r]);
      const float mn = fmaxf(mrow[r], rm);
      const float sc = __expf(mrow[r] - mn);  // exp(-inf)=0 on first tile
      mrow[r] = mn;
      const float p = __expf(E[r] - mn);
      E[r] = p;
      lrow[r] = lrow[r] * sc + rsum16(p);
      scr[r] = sc;
    }
#pragma unroll
    for (int ct = 0; ct < 4; ++ct)
#pragma unroll
      for (int r = 0; r < 8; ++r) acc[ct][r] *= scr[r];

    // stage P (C layout) -> LDS so it can be re-read in A layout
#pragma unroll
    for (int r = 0; r < 8; ++r) myp[(r + 8 * hh) * 20 + l16] = E[r];

    // O += P(16x16) * V^T(16x64): 4 k-steps x 4 channel tiles
#pragma unroll
    for (int kk2 = 0; kk2 < 4; ++kk2) {
      v2f a2 = *(const v2f*)(myp + l16 * 20 + kk2 * 4 + 2 * hh);
      const float* vt = vb + jt * 16 + kk2 * 4 + 2 * hh;
#pragma unroll
      for (int ct = 0; ct < 4; ++ct)
        acc[ct] = wmma_f32(a2, *(const v2f*)(vt + (size_t)(ct * 16 + l16) * NSTR),
                           acc[ct]);
    }
  }

  // epilogue: divide by softmax sum, add residual, write the owned pixel
#pragma unroll
  for (int r = 0; r < 8; ++r) {
    const float inv = 1.0f / lrow[r];
    const int m  = r + 8 * hh;          // query row of this VGPR in this half
    const int oi = qt * 16 + m;
    const int lr = oi >> 5, lc = oi & 31;
    const size_t poff = (size_t)(wi * 32 + lr) * HW + (wj * 32 + lc);
#pragma unroll
    for (int ct = 0; ct < 4; ++ct) {
      const size_t off = (size_t)(ct * 16 + l16) * IMG + poff;
      dst[off] = acc[ct][r] * inv + src[off];
    }
  }
}

// ---------------------------------------------------------------------------
extern "C" void kernel_launch(void* const* d_in, const int* in_sizes, int n_in,
                              void* d_out, int out_size, void* d_ws,
                              size_t ws_size, hipStream_t stream) {
  (void)in_sizes; (void)n_in; (void)out_size; (void)ws_size;
  const float* x = (const float*)d_in[0];
  const float* Wq[2] = {(const float*)d_in[1], (const float*)d_in[4]};
  const float* Wk[2] = {(const float*)d_in[2], (const float*)d_in[5]};
  const float* Wv[2] = {(const float*)d_in[3], (const float*)d_in[6]};
  const float* bq[2] = {(const float*)d_in[7], (const float*)d_in[10]};
  const float* bk[2] = {(const float*)d_in[8], (const float*)d_in[11]};
  const float* bv[2] = {(const float*)d_in[9], (const float*)d_in[12]};
  float* out = (float*)d_out;

  float* qbuf = (float*)d_ws;                       // 16*1024*64 f32
  float* kbuf = qbuf + (size_t)16 * QOWN * NCH;     // 16*2304*64 f32
  float* vbuf = kbuf + (size_t)16 * NSTR * NCH;     // 16*2304*64 f32

  for (int layer = 0; layer < 2; ++layer) {
    const float* srcimg = (layer == 0) ? x : out;  // layer2 reads layer1 out
    for (int b = 0; b < 8; ++b) {
      const float* sb = srcimg + (size_t)b * NCH * IMG;
      float* ob = out + (size_t)b * NCH * IMG;
      qkv_kernel<<<dim3(16 * 144), dim3(32), 0, stream>>>(
          sb, Wq[layer], Wk[layer], Wv[layer], bq[layer], bk[layer], bv[layer],
          qbuf, kbuf, vbuf);
      attn_kernel<<<dim3(128), dim3(256), 0, stream>>>(qbuf, kbuf, vbuf, sb,
                                                       ob);
    }
  }
}